// GATv2_onehop_29618094474249
// MI455X (gfx1250) — compile-verified
//
#include <hip/hip_runtime.h>
#include <hip/hip_bf16.h>

typedef __attribute__((ext_vector_type(2))) float v2f;
typedef __attribute__((ext_vector_type(8))) float v8f;

#define CH 64           // channels per head
#define HEADS 4
#define HC 256          // HEADS*CH
#define NEG_SLOPE 0.2f

// ---------------------------------------------------------------------------
// Init workspace: accum[N*256]=0, smax[N*4]=-inf, den[N*4]=0
// ---------------------------------------------------------------------------
__global__ void gat_init_kernel(float* __restrict__ accum,
                                float* __restrict__ smax,
                                float* __restrict__ den, int N) {
    int idx = blockIdx.x * blockDim.x + threadIdx.x;
    int total = N * HC;
    if (idx < total) accum[idx] = 0.0f;
    if (idx < N * HEADS) {
        smax[idx] = -__builtin_inff();
        den[idx]  = 0.0f;
    }
}

// ---------------------------------------------------------------------------
// Projection: xl = x @ W_l + b_l ; xr = x @ W_r + b_r
// One block = 16 rows of x staged in LDS; 8 waves x 4 col-tiles = all 32
// 16-wide column tiles of the concatenated [W_l | W_r] output (2 * 256 cols).
// Uses V_WMMA_F32_16X16X4_F32, accumulating K=64 in 16 steps.
// A 16x4 f32 layout: lane m=lane&15; VGPR0/1 = K = koff, koff+1 where
// koff = 2*(lane>=16). B 4x16 mirrored with N on lanes. D: row = v + 8*(lane>=16).
// ---------------------------------------------------------------------------
__global__ void gat_proj_kernel(const float* __restrict__ x,
                                const float* __restrict__ Wl,
                                const float* __restrict__ bl,
                                const float* __restrict__ Wr,
                                const float* __restrict__ br,
                                float* __restrict__ xl,
                                float* __restrict__ xr, int N) {
    __shared__ float As[16 * CH];
    const int row0 = blockIdx.x * 16;
    const int tid  = threadIdx.x;          // 256 threads = 8 waves

    // stage 16x64 tile of x (N is a multiple of 16 -> no guards, EXEC all 1s)
    for (int i = tid; i < 16 * CH; i += 256) {
        int r = i >> 6, c = i & 63;
        As[i] = x[(size_t)(row0 + r) * CH + c];
    }
    __syncthreads();

    const int wave = tid >> 5;
    const int lane = tid & 31;
    const int m    = lane & 15;
    const int koff = (lane >> 4) << 1;     // 0 or 2
    const int mrow = (lane >> 4) << 3;     // 0 or 8 (D-matrix row base)

    for (int t = 0; t < 4; ++t) {
        const int colTile = wave * 4 + t;                 // 0..31
        const bool left   = (colTile < 16);
        const float* __restrict__ W = left ? Wl : Wr;
        const float* __restrict__ b = left ? bl : br;
        const int n = ((colTile & 15) << 4) + m;          // output column 0..255

        v8f acc;
        const float bv = b[n];
        #pragma unroll
        for (int v = 0; v < 8; ++v) acc[v] = bv;          // bias folded into C

        #pragma unroll
        for (int k0 = 0; k0 < CH; k0 += 4) {
            v2f a, bf;
            a.x  = As[m * CH + k0 + koff];
            a.y  = As[m * CH + k0 + koff + 1];
            bf.x = W[(size_t)(k0 + koff) * HC + n];
            bf.y = W[(size_t)(k0 + koff + 1) * HC + n];
            acc = __builtin_amdgcn_wmma_f32_16x16x4_f32(
                false, a, false, bf, (short)0, acc, false, false);
        }

        float* __restrict__ out = left ? xl : xr;
        #pragma unroll
        for (int v = 0; v < 8; ++v)
            out[(size_t)(row0 + mrow + v) * HC + n] = acc[v];
    }
}

// ---------------------------------------------------------------------------
// Pass A: per-edge GATv2 scores + segment max via f32 atomic max (L2 resident).
// One wave per edge; lane owns 8 contiguous channels (2 x float4), so each
// lane touches exactly one head (h = lane>>3); 3-step shfl_xor reduction.
// ---------------------------------------------------------------------------
__global__ void gat_score_kernel(const float* __restrict__ xl,
                                 const float* __restrict__ xr,
                                 const float* __restrict__ att,
                                 const int* __restrict__ edges,
                                 float* __restrict__ score,
                                 float* __restrict__ smax,
                                 int E, int Etot) {
    const int widx = (int)((blockIdx.x * (size_t)blockDim.x + threadIdx.x) >> 5);
    const int lane = threadIdx.x & 31;
    if (widx >= Etot) return;

    int src, dst;
    if (widx < E) { src = edges[widx]; dst = edges[E + widx]; }
    else          { src = widx - E;    dst = src; }             // self loops

    const float4* __restrict__ pl = (const float4*)(xl + (size_t)src * HC);
    const float4* __restrict__ pr = (const float4*)(xr + (size_t)dst * HC);
    const float4* __restrict__ pa = (const float4*)att;          // [4][64]

    float s = 0.0f;
    #pragma unroll
    for (int i = 0; i < 2; ++i) {
        const int q = lane * 2 + i;
        float4 a = pl[q], b = pr[q], w = pa[q];
        float vx = a.x + b.x; vx = vx > 0.0f ? vx : NEG_SLOPE * vx;
        float vy = a.y + b.y; vy = vy > 0.0f ? vy : NEG_SLOPE * vy;
        float vz = a.z + b.z; vz = vz > 0.0f ? vz : NEG_SLOPE * vz;
        float vw = a.w + b.w; vw = vw > 0.0f ? vw : NEG_SLOPE * vw;
        s += w.x * vx + w.y * vy + w.z * vz + w.w * vw;
    }
    // reduce the 8 lanes belonging to one head
    s += __shfl_xor(s, 1, 32);
    s += __shfl_xor(s, 2, 32);
    s += __shfl_xor(s, 4, 32);

    if ((lane & 7) == 0) {
        const int h = lane >> 3;
        score[(size_t)widx * HEADS + h] = s;
        __hip_atomic_fetch_max(&smax[dst * HEADS + h], s,
                               __ATOMIC_RELAXED, __HIP_MEMORY_SCOPE_AGENT);
    }
}

// ---------------------------------------------------------------------------
// Pass B: ex = exp(score - smax[dst]); den[dst] += ex (one thread per edge*head)
// ---------------------------------------------------------------------------
__global__ void gat_exp_kernel(const int* __restrict__ edges,
                               const float* __restrict__ smax,
                               float* __restrict__ score,   // in: score, out: ex
                               float* __restrict__ den,
                               int E, int Etot) {
    const int idx = blockIdx.x * blockDim.x + threadIdx.x;
    if (idx >= Etot * HEADS) return;
    const int e = idx >> 2, h = idx & 3;
    const int dst = (e < E) ? edges[E + e] : (e - E);
    const float ex = __expf(score[idx] - smax[dst * HEADS + h]);
    score[idx] = ex;
    __hip_atomic_fetch_add(&den[dst * HEADS + h], ex,
                           __ATOMIC_RELAXED, __HIP_MEMORY_SCOPE_AGENT);
}

// ---------------------------------------------------------------------------
// Pass C: accum[dst] += (ex/den[dst]) * xl[src]  (wave per edge, 8 ch/lane)
// ---------------------------------------------------------------------------
__global__ void gat_aggr_kernel(const float* __restrict__ xl,
                                const int* __restrict__ edges,
                                const float* __restrict__ ex,
                                const float* __restrict__ den,
                                float* __restrict__ accum,
                                int E, int Etot) {
    const int widx = (int)((blockIdx.x * (size_t)blockDim.x + threadIdx.x) >> 5);
    const int lane = threadIdx.x & 31;
    if (widx >= Etot) return;

    int src, dst;
    if (widx < E) { src = edges[widx]; dst = edges[E + widx]; }
    else          { src = widx - E;    dst = src; }

    const int h = lane >> 3;
    const float alpha = ex[(size_t)widx * HEADS + h] / den[dst * HEADS + h];

    const float4* __restrict__ pl = (const float4*)(xl + (size_t)src * HC);
    float* __restrict__ pa = accum + (size_t)dst * HC;

    #pragma unroll
    for (int i = 0; i < 2; ++i) {
        const int q = lane * 2 + i;
        float4 v = pl[q];
        const int base = q << 2;
        __hip_atomic_fetch_add(&pa[base + 0], alpha * v.x, __ATOMIC_RELAXED, __HIP_MEMORY_SCOPE_AGENT);
        __hip_atomic_fetch_add(&pa[base + 1], alpha * v.y, __ATOMIC_RELAXED, __HIP_MEMORY_SCOPE_AGENT);
        __hip_atomic_fetch_add(&pa[base + 2], alpha * v.z, __ATOMIC_RELAXED, __HIP_MEMORY_SCOPE_AGENT);
        __hip_atomic_fetch_add(&pa[base + 3], alpha * v.w, __ATOMIC_RELAXED, __HIP_MEMORY_SCOPE_AGENT);
    }
}

// ---------------------------------------------------------------------------
// Finalize: out[n][c] = mean_h accum[n][h*64+c] + bias[c]
// ---------------------------------------------------------------------------
__global__ void gat_final_kernel(const float* __restrict__ accum,
                                 const float* __restrict__ bias,
                                 float* __restrict__ out, int N) {
    const int idx = blockIdx.x * blockDim.x + threadIdx.x;
    if (idx >= N * CH) return;
    const int n = idx >> 6, c = idx & 63;
    const float* __restrict__ a = accum + (size_t)n * HC;
    out[idx] = 0.25f * (a[c] + a[CH + c] + a[2 * CH + c] + a[3 * CH + c]) + bias[c];
}

// ---------------------------------------------------------------------------
extern "C" void kernel_launch(void* const* d_in, const int* in_sizes, int n_in,
                              void* d_out, int out_size, void* d_ws, size_t ws_size,
                              hipStream_t stream) {
    const float* x    = (const float*)d_in[0];
    const float* Wl   = (const float*)d_in[1];
    const float* bl   = (const float*)d_in[2];
    const float* Wr   = (const float*)d_in[3];
    const float* br   = (const float*)d_in[4];
    const float* att  = (const float*)d_in[5];
    const float* bias = (const float*)d_in[6];
    const int*   edges = (const int*)d_in[7];

    const int N    = in_sizes[0] / CH;   // 50000
    const int E    = in_sizes[7] / 2;    // 800000
    const int Etot = E + N;              // edges + self loops

    // workspace layout (f32): xl[N*256] xr[N*256] accum[N*256] smax[N*4] den[N*4] score[Etot*4]
    float* xl    = (float*)d_ws;
    float* xr    = xl + (size_t)N * HC;
    float* accum = xr + (size_t)N * HC;
    float* smax  = accum + (size_t)N * HC;
    float* den   = smax + (size_t)N * HEADS;
    float* score = den + (size_t)N * HEADS;

    const int B = 256;
    gat_init_kernel<<<(N * HC + B - 1) / B, B, 0, stream>>>(accum, smax, den, N);
    gat_proj_kernel<<<N / 16, B, 0, stream>>>(x, Wl, bl, Wr, br, xl, xr, N);

    const long long waveThreads = (long long)Etot * 32;
    gat_score_kernel<<<(int)((waveThreads + B - 1) / B), B, 0, stream>>>(
        xl, xr, att, edges, score, smax, E, Etot);
    gat_exp_kernel<<<(Etot * HEADS + B - 1) / B, B, 0, stream>>>(
        edges, smax, score, den, E, Etot);
    gat_aggr_kernel<<<(int)((waveThreads + B - 1) / B), B, 0, stream>>>(
        xl, edges, score, den, accum, E, Etot);
    gat_final_kernel<<<(N * CH + B - 1) / B, B, 0, stream>>>(
        accum, bias, (float*)d_out, N);
}